// GatedImageAttention_137438953727
// MI455X (gfx1250) — compile-verified
//
#include <hip/hip_runtime.h>

#define B_ 32
#define S_ 2048
#define F_ 768
#define H_ 8

typedef __attribute__((ext_vector_type(2))) float v2f;
typedef __attribute__((ext_vector_type(8))) float v8f;
typedef __attribute__((ext_vector_type(4))) unsigned int u32x4;
typedef __attribute__((ext_vector_type(4))) int i32x4;
typedef __attribute__((ext_vector_type(8))) int i32x8;

// TDM tile geometry for the pooled kernel
#define CH_S 64          // s rows per chunk
#define CH_F 128         // f columns per workgroup
#define LDS_PITCH 132    // floats per LDS row (128 + 4 pad; matches TDM pad cfg)

// ---------------------------------------------------------------------------
// Issue one TDM 2D tile DMA: global (CH_S x CH_F fp32 tile, row stride F_)
// -> LDS with 4-DWORD padding after every 128 DWORDs (row pitch = 132 floats).
// D# layout per CDNA5 ISA.  Wave-level op (EXEC ignored).
// ---------------------------------------------------------------------------
__device__ __forceinline__ void tdm_load_tile(const float* gptr, unsigned lds_off) {
  unsigned long long ga = (unsigned long long)(uintptr_t)gptr;
  u32x4 g0;
  g0.x = 1u;                                                  // count=1, user D#
  g0.y = lds_off;                                             // lds_addr (bytes)
  g0.z = (unsigned)(ga & 0xFFFFFFFFull);                      // global_addr[31:0]
  g0.w = (unsigned)((ga >> 32) & 0x1FFFFFFull) | (2u << 30);  // [56:32] | type=2

  const unsigned td0   = F_;    // tensor_dim0 (elements along f)
  const unsigned td1   = S_;    // tensor_dim1 (elements along s)
  const unsigned tile0 = CH_F;  // tile_dim0
  const unsigned tile1 = CH_S;  // tile_dim1
  const unsigned str0  = F_;    // tensor_dim0_stride

  i32x8 g1;
  g1[0] = (int)((2u << 16)     // data_size = 4 bytes
              | (1u << 20)     // pad_enable
              | (6u << 22)     // pad_interval: pad after every 128 DWORDs
              | (3u << 25));   // pad_amount: 4 DWORDs (16 B)
  g1[1] = (int)((td0 & 0xFFFFu) << 16);                     // dim0 lo16 @ [31:16]
  g1[2] = (int)((td0 >> 16) | ((td1 & 0xFFFFu) << 16));     // dim0 hi | dim1 lo
  g1[3] = (int)((td1 >> 16) | (tile0 << 16));               // dim1 hi | tile_dim0
  g1[4] = (int)tile1;                                       // tile_dim1 | tile_dim2=0
  g1[5] = (int)str0;                                        // dim0_stride[31:0]
  g1[6] = 0;                                                // stride0 hi | stride1 lo
  g1[7] = 0;                                                // stride1 hi (unused, 2D)

  i32x4 gz = {0, 0, 0, 0};
#if __clang_major__ >= 23
  i32x8 gz8 = {0, 0, 0, 0, 0, 0, 0, 0};
  __builtin_amdgcn_tensor_load_to_lds(g0, g1, gz, gz, gz8, 0);
#else
  __builtin_amdgcn_tensor_load_to_lds(g0, g1, gz, gz, 0);
#endif
}

// ---------------------------------------------------------------------------
// Kernel 1: att/gate logits.  One wave per 16 rows of x, K = 768 WMMA chain.
// B matrix packs [W_att ; W_gate] as 16 columns.  Epilogue stores a single
// interleaved [row][16] buffer (cols 0..7 att, 8..15 gate): branch-free,
// two coalesced 64B segments per accumulator register.
// ---------------------------------------------------------------------------
__global__ __launch_bounds__(32) void logits_wmma(
    const float* __restrict__ x, const float* __restrict__ Wa,
    const float* __restrict__ ba, const float* __restrict__ Wg,
    const float* __restrict__ bg, float* __restrict__ logitsC) {
  const int lane = threadIdx.x;
  const int m = lane & 15;
  const int half = lane >> 4;
  const long row0 = (long)blockIdx.x * 16;

  const float* xrow = x + (row0 + m) * F_;
  const float* wrow = (m < 8) ? (Wa + (size_t)m * F_) : (Wg + (size_t)(m - 8) * F_);

  v8f acc = {};
#pragma unroll 8
  for (int k0 = 0; k0 < F_; k0 += 4) {
    v2f a = *(const v2f*)(xrow + k0 + 2 * half);
    v2f b = *(const v2f*)(wrow + k0 + 2 * half);
    acc = __builtin_amdgcn_wmma_f32_16x16x4_f32(false, a, false, b,
                                                (short)0, acc, false, false);
  }

  const int n = m;  // D column: 0..7 att head, 8..15 gate head
  const float bias = (n < 8) ? ba[n] : bg[n - 8];
#pragma unroll
  for (int j = 0; j < 8; ++j) {
    const long row = row0 + j + 8 * half;
    logitsC[row * 16 + n] = acc[j] + bias;
  }
}

// ---------------------------------------------------------------------------
// Kernel 2: per-(b,h) softmax over S, times sigmoid(gate).  LDS reductions.
// ---------------------------------------------------------------------------
__global__ __launch_bounds__(256) void softmax_gate(
    const float* __restrict__ logitsC, float* __restrict__ w) {
  const int bh = blockIdx.x;
  const int b = bh >> 3, h = bh & 7;
  const size_t rbase = (size_t)b * S_;
  const int tid = threadIdx.x;
  __shared__ float red[256];

  float lmax = -3.4e38f;
  for (int s = tid; s < S_; s += 256)
    lmax = fmaxf(lmax, logitsC[(rbase + s) * 16 + h]);
  red[tid] = lmax;
  __syncthreads();
  for (int off = 128; off > 0; off >>= 1) {
    if (tid < off) red[tid] = fmaxf(red[tid], red[tid + off]);
    __syncthreads();
  }
  const float gmax = red[0];
  __syncthreads();

  float lsum = 0.f;
  for (int s = tid; s < S_; s += 256)
    lsum += __expf(logitsC[(rbase + s) * 16 + h] - gmax);
  red[tid] = lsum;
  __syncthreads();
  for (int off = 128; off > 0; off >>= 1) {
    if (tid < off) red[tid] += red[tid + off];
    __syncthreads();
  }
  const float inv = 1.0f / red[0];

  for (int s = tid; s < S_; s += 256) {
    float e  = __expf(logitsC[(rbase + s) * 16 + h] - gmax) * inv;
    float g  = logitsC[(rbase + s) * 16 + 8 + h];
    float sg = 1.0f / (1.0f + __expf(-g));
    w[(rbase + s) * H_ + h] = e * sg;
  }
}

// ---------------------------------------------------------------------------
// Kernel 3: pooled[b,f,h] = sum_s x[b,s,f] * w[b,s,h].
// Workgroup = 8 waves, one batch, 128-wide f block.  x tiles (64 s x 128 f)
// are DMA'd into LDS by the Tensor Data Mover (double-buffered, TENSORcnt),
// with TDM padding giving a 132-float LDS pitch (bank-conflict-free for the
// two WMMA half-wave K rows).  Each wave owns one 16-wide f tile.
// ---------------------------------------------------------------------------
__global__ __launch_bounds__(256) void pooled_wmma(
    const float* __restrict__ x, const float* __restrict__ w,
    float* __restrict__ pooled) {
  __shared__ float xs[2][CH_S * LDS_PITCH];

  const int tid = threadIdx.x;
  const int lane = tid & 31;
  const int wave = tid >> 5;            // 0..7
  const int n = lane & 15;              // A row (f) and B/D column (head)
  const int half = lane >> 4;
  const int b  = blockIdx.x / (F_ / CH_F);
  const int f0 = (blockIdx.x % (F_ / CH_F)) * CH_F;
  const int fl = wave * 16;             // f offset inside LDS tile

  const float* xb = x + (size_t)b * S_ * F_ + f0;   // tile column origin
  const float* wb = w + (size_t)b * S_ * H_;
  const float sel = (n < 8) ? 1.0f : 0.0f;          // zero-pad heads 8..15
  const int hc = n & 7;

  v8f acc = {};
  const int nchunk = S_ / CH_S;         // 32

  if (wave == 0)
    tdm_load_tile(xb, (unsigned)(uintptr_t)&xs[0][0]);

  for (int c = 0; c < nchunk; ++c) {
    __syncthreads();                    // prev buffer fully consumed
    if (wave == 0) {
      if (c + 1 < nchunk) {
        tdm_load_tile(xb + (size_t)(c + 1) * CH_S * F_,
                      (unsigned)(uintptr_t)&xs[(c + 1) & 1][0]);
        __builtin_amdgcn_s_wait_tensorcnt((short)1);  // chunk c landed
      } else {
        __builtin_amdgcn_s_wait_tensorcnt((short)0);
      }
    }
    __syncthreads();                    // publish chunk c to all waves
    const float* xt = &xs[c & 1][0];
    const int s0 = c * CH_S;
#pragma unroll 4
    for (int kk = 0; kk < CH_S; kk += 4) {
      const int k = kk + 2 * half;
      v2f a, bb;
      a.x  = xt[k * LDS_PITCH + fl + n];
      a.y  = xt[(k + 1) * LDS_PITCH + fl + n];
      bb.x = wb[(size_t)(s0 + k) * H_ + hc] * sel;
      bb.y = wb[(size_t)(s0 + k + 1) * H_ + hc] * sel;
      acc = __builtin_amdgcn_wmma_f32_16x16x4_f32(false, a, false, bb,
                                                  (short)0, acc, false, false);
    }
  }

  if (n < 8) {
#pragma unroll
    for (int j = 0; j < 8; ++j) {
      const int f = f0 + fl + j + 8 * half;
      pooled[((size_t)b * F_ + f) * H_ + n] = acc[j];
    }
  }
}

// ---------------------------------------------------------------------------
// Kernel 4: transpose W_out [768 x 6144] -> W_outT [6144 x 768] (LDS tile)
// so the final GEMM's B loads are coalesced.
// ---------------------------------------------------------------------------
__global__ __launch_bounds__(1024) void transpose_wout(
    const float* __restrict__ Wo, float* __restrict__ WoT) {
  __shared__ float t[32][33];
  const int k0 = blockIdx.x * 32;
  const int n0 = blockIdx.y * 32;
  const int tx = threadIdx.x, ty = threadIdx.y;
  t[ty][tx] = Wo[(size_t)(n0 + ty) * (F_ * H_) + k0 + tx];
  __syncthreads();
  WoT[(size_t)(k0 + ty) * F_ + n0 + tx] = t[tx][ty];
}

// ---------------------------------------------------------------------------
// Kernel 5: out[32,768] = pooled[32,6144] @ W_outT + b_out, WMMA, K = 6144.
// ---------------------------------------------------------------------------
__global__ __launch_bounds__(32) void out_gemm_wmma(
    const float* __restrict__ pooled, const float* __restrict__ WoT,
    const float* __restrict__ bo, float* __restrict__ out) {
  const int lane = threadIdx.x;
  const int n = lane & 15;
  const int half = lane >> 4;
  const int mt = blockIdx.x / (F_ / 16);   // 0..1
  const int n0 = (blockIdx.x % (F_ / 16)) * 16;
  const int K = F_ * H_;                   // 6144

  const float* prow = pooled + (size_t)(mt * 16 + n) * K;

  v8f acc = {};
#pragma unroll 4
  for (int k0 = 0; k0 < K; k0 += 4) {
    const int k = k0 + 2 * half;
    v2f a = *(const v2f*)(prow + k);
    v2f bb;
    bb.x = WoT[(size_t)k * F_ + n0 + n];
    bb.y = WoT[(size_t)(k + 1) * F_ + n0 + n];
    acc = __builtin_amdgcn_wmma_f32_16x16x4_f32(false, a, false, bb,
                                                (short)0, acc, false, false);
  }

  const float bias = bo[n0 + n];
#pragma unroll
  for (int j = 0; j < 8; ++j) {
    const int m = mt * 16 + j + 8 * half;
    out[(size_t)m * F_ + n0 + n] = acc[j] + bias;
  }
}

// ---------------------------------------------------------------------------
extern "C" void kernel_launch(void* const* d_in, const int* in_sizes, int n_in,
                              void* d_out, int out_size, void* d_ws, size_t ws_size,
                              hipStream_t stream) {
  const float* x  = (const float*)d_in[0];
  const float* Wa = (const float*)d_in[1];
  const float* ba = (const float*)d_in[2];
  const float* Wg = (const float*)d_in[3];
  const float* bg = (const float*)d_in[4];
  const float* Wo = (const float*)d_in[5];
  const float* bo = (const float*)d_in[6];
  float* out = (float*)d_out;

  float* ws = (float*)d_ws;
  const size_t BS = (size_t)B_ * S_;
  float* logitsC = ws;                            // BS*16     = 1048576 floats
  float* w       = logitsC + BS * 16;             // BS*H      = 524288
  float* pooled  = w + BS * H_;                   // B*F*H     = 196608
  float* WoT     = pooled + (size_t)B_ * F_ * H_; // 6144*768  = 4718592

  logits_wmma<<<(B_ * S_) / 16, 32, 0, stream>>>(x, Wa, ba, Wg, bg, logitsC);
  softmax_gate<<<B_ * H_, 256, 0, stream>>>(logitsC, w);
  pooled_wmma<<<B_ * (F_ / CH_F), 256, 0, stream>>>(x, w, pooled);
  transpose_wout<<<dim3((F_ * H_) / 32, F_ / 32), dim3(32, 32), 0, stream>>>(Wo, WoT);
  out_gemm_wmma<<<2 * (F_ / 16), 32, 0, stream>>>(pooled, WoT, bo, out);
}